// Attention_2293512536207
// MI455X (gfx1250) — compile-verified
//
#include <hip/hip_runtime.h>
#include <hip/hip_bf16.h>

// Problem constants (from the reference)
#define B_DIM 16384
#define S_DIM 512
#define P_DIM 1024
#define K_DIM 8
#define D_DIM 64
#define H_DIM 128

typedef __attribute__((ext_vector_type(2))) float v2f;
typedef __attribute__((ext_vector_type(8))) float v8f;

// ---------------------------------------------------------------------------
// Kernel 1: projP[p,h] = sum_d plm[p,d]*W[h,d];  projS[s,h] = sum_d skl[s,d]*U[h,d]
// One wave computes one 16x16 (M x H) f32 tile via 16 chained
// v_wmma_f32_16x16x4_f32 (K = D = 64 in steps of 4). Full fp32 precision,
// no conversion VALU. 8 waves / 256-thread block.
//
// 32-bit A 16x4 layout (ISA 7.12.2): lanes 0-15 hold M=lane, VGPR0=K0,VGPR1=K1;
// lanes 16-31 hold M=lane-16, VGPR0=K2,VGPR1=K3. B (4x16) mirrored on N.
// ---------------------------------------------------------------------------
__global__ __launch_bounds__(256) void proj_wmma_kernel(
    const float* __restrict__ plm, const float* __restrict__ W,
    const float* __restrict__ skl, const float* __restrict__ U,
    float* __restrict__ projP, float* __restrict__ projS)
{
    const int wave = threadIdx.x >> 5;
    const int lane = threadIdx.x & 31;
    const int tile = blockIdx.x * 8 + wave;       // 0..767
    const int mt   = tile >> 3;                   // 0..95
    const int nt   = tile & 7;                    // 0..7

    const float* E;  const float* M;  float* out;  int m0;
    if (mt < (P_DIM / 16)) { E = plm; M = W; out = projP; m0 = mt * 16; }
    else                   { E = skl; M = U; out = projS; m0 = (mt - P_DIM / 16) * 16; }

    const int half = lane >> 4;       // 0 or 1
    const int l16  = lane & 15;
    const int n0   = nt * 16;
    const int kp   = 2 * half;        // K-pair base within each K=4 step

    const float* arow = E + (m0 + l16) * D_DIM;   // A[m,k] = E[m0+m, k]
    const float* brow = M + (n0 + l16) * D_DIM;   // B[k,n] = M[n0+n, k]

    v8f c = {};
    #pragma unroll
    for (int k0 = 0; k0 < D_DIM; k0 += 4) {
        v2f a, b;
        a[0] = arow[k0 + kp];
        a[1] = arow[k0 + kp + 1];
        b[0] = brow[k0 + kp];
        b[1] = brow[k0 + kp + 1];
        // (neg_a, A, neg_b, B, c_mod, C, reuse_a, reuse_b)
        c = __builtin_amdgcn_wmma_f32_16x16x4_f32(
                false, a, false, b, (short)0, c, false, false);
    }

    // C/D layout: VGPR r, lanes 0-15: M=r, N=lane; lanes 16-31: M=r+8, N=lane-16
    #pragma unroll
    for (int r = 0; r < 8; ++r) {
        int m = m0 + r + 8 * half;
        out[m * H_DIM + n0 + l16] = c[r];
    }
}

// ---------------------------------------------------------------------------
// Kernel 2: scores[p,k] = sum_h v[h]*tanh(projP[p,h] + projS[g[p,k],h]);
// att = softmax_k(scores). One wave per p (wave32); each lane owns 4 h values,
// wave-allreduce via __shfl_xor. 8 waves / block, grid = P/8 = 128 blocks.
// ---------------------------------------------------------------------------
__global__ __launch_bounds__(256) void score_softmax_kernel(
    const float* __restrict__ projP, const float* __restrict__ projS,
    const float* __restrict__ vT, const int* __restrict__ gidx,
    float* __restrict__ att)
{
    const int wave = threadIdx.x >> 5;
    const int lane = threadIdx.x & 31;
    const int p    = blockIdx.x * 8 + wave;       // 0..1023

    float vp[4], vv[4];
    #pragma unroll
    for (int j = 0; j < 4; ++j) {
        int h = lane + 32 * j;
        vp[j] = projP[p * H_DIM + h];
        vv[j] = vT[h];
    }

    float sc[K_DIM];
    #pragma unroll
    for (int k = 0; k < K_DIM; ++k) {
        int s = gidx[p * K_DIM + k];
        float partial = 0.0f;
        #pragma unroll
        for (int j = 0; j < 4; ++j) {
            int h = lane + 32 * j;
            partial += vv[j] * tanhf(vp[j] + projS[s * H_DIM + h]);
        }
        #pragma unroll
        for (int off = 16; off >= 1; off >>= 1)
            partial += __shfl_xor(partial, off, 32);
        sc[k] = partial;                           // all lanes hold the sum
    }

    float m = sc[0];
    #pragma unroll
    for (int k = 1; k < K_DIM; ++k) m = fmaxf(m, sc[k]);
    float e[K_DIM], sum = 0.0f;
    #pragma unroll
    for (int k = 0; k < K_DIM; ++k) { e[k] = expf(sc[k] - m); sum += e[k]; }
    float inv = 1.0f / sum;
    if (lane < K_DIM) att[p * K_DIM + lane] = e[lane] * inv;
}

// ---------------------------------------------------------------------------
// Kernel 3 (bandwidth-bound stage, ~160 MB HBM -> ~7 us floor):
// out[b,p] = mask[b,p] * sum_k skl_pfc[b, g[p,k]] * att[p,k]
// Block = 256 threads, tile = 256 p x 64 b. Per b, stage the 2 KB skl_pfc row
// in LDS so the K=8 random gathers hit LDS; mask read + out write coalesced.
// ---------------------------------------------------------------------------
#define PT 256
#define BT 64
__global__ __launch_bounds__(256) void out_gather_kernel(
    const float* __restrict__ pfc, const float* __restrict__ mask,
    const int* __restrict__ gidx, const float* __restrict__ att,
    float* __restrict__ out)
{
    __shared__ float row_s[S_DIM];

    const int t  = threadIdx.x;
    const int p0 = blockIdx.x * PT;
    const int b0 = blockIdx.y * BT;
    const int p  = p0 + t;

    float a_r[K_DIM];
    int   g_r[K_DIM];
    #pragma unroll
    for (int k = 0; k < K_DIM; ++k) {
        a_r[k] = att[p * K_DIM + k];
        g_r[k] = gidx[p * K_DIM + k];
    }

    for (int bi = 0; bi < BT; ++bi) {
        const int b = b0 + bi;
        // cooperative row load: 512 floats / 256 threads = 2 each, coalesced
        row_s[t]       = pfc[b * S_DIM + t];
        row_s[t + 256] = pfc[b * S_DIM + t + 256];
        __syncthreads();

        float acc = 0.0f;
        #pragma unroll
        for (int k = 0; k < K_DIM; ++k)
            acc += row_s[g_r[k]] * a_r[k];

        const int o = b * P_DIM + p;
        out[o] = acc * mask[o];
        __syncthreads();   // protect row_s before next iteration's overwrite
    }
}

// ---------------------------------------------------------------------------
// Launch
// ---------------------------------------------------------------------------
extern "C" void kernel_launch(void* const* d_in, const int* in_sizes, int n_in,
                              void* d_out, int out_size, void* d_ws, size_t ws_size,
                              hipStream_t stream) {
    const float* skl_pfc = (const float*)d_in[0];   // [B,S]
    const float* mask    = (const float*)d_in[1];   // [B,P]
    const float* skl_emd = (const float*)d_in[2];   // [S,D]
    const float* plm_emd = (const float*)d_in[3];   // [P,D]
    const float* W       = (const float*)d_in[4];   // [H,D]
    const float* U       = (const float*)d_in[5];   // [H,D]
    const float* vT      = (const float*)d_in[6];   // [1,H]
    const int*   gidx    = (const int*)d_in[7];     // [P,K]
    float* out = (float*)d_out;                     // [B,P]

    float* projP = (float*)d_ws;                    // P*H floats (512 KB)
    float* projS = projP + P_DIM * H_DIM;           // S*H floats (256 KB)
    float* att   = projS + S_DIM * H_DIM;           // P*K floats ( 32 KB)

    // 768 tiles / 8 waves per block = 96 blocks
    proj_wmma_kernel<<<dim3(96), 256, 0, stream>>>(plm_emd, W, skl_emd, U, projP, projS);
    score_softmax_kernel<<<dim3(P_DIM / 8), 256, 0, stream>>>(projP, projS, vT, gidx, att);
    out_gather_kernel<<<dim3(P_DIM / PT, B_DIM / BT), 256, 0, stream>>>(
        skl_pfc, mask, gidx, att, out);
}